// SlaterJastrow_85066122264744
// MI455X (gfx1250) — compile-verified
//
#include <hip/hip_runtime.h>
#include <hip/hip_bf16.h>

typedef __attribute__((ext_vector_type(2))) float v2f;
typedef __attribute__((ext_vector_type(8))) float v8f;

#define BATCH   32768
#define N_ORB   64
#define NF      32
#define NV      128   // N_SPIN * N_ORB
#define PI_F    3.14159265358979323846f

// ---------------------------------------------------------------------------
// Kernel 1: densify strictly-lower-triangular W into a WMMA-B-fragment
// friendly packed layout:
//   Wpack[kk][j] = { W[2kk][j], W[2kk+1][j] }   (kk = 0..63, j = 0..127)
// so a lane's whole 4x16 B fragment column-pair is one 8-byte load.
// W[i][j] = jk[i*(i-1)/2 + j] for j < i, else 0.
// ---------------------------------------------------------------------------
__global__ __launch_bounds__(128) void build_w_kernel(const float* __restrict__ jk,
                                                      float* __restrict__ Wp) {
    const int kk = blockIdx.x;    // 0..63  (row pair 2kk, 2kk+1)
    const int j  = threadIdx.x;   // 0..127 col
    const int i0 = 2 * kk;
    const int i1 = 2 * kk + 1;
    float w0 = (j < i0) ? jk[(i0 * (i0 - 1)) / 2 + j] : 0.0f;
    float w1 = (j < i1) ? jk[(i1 * (i1 - 1)) / 2 + j] : 0.0f;
    v2f p; p.x = w0; p.y = w1;
    *(v2f*)(Wp + ((size_t)kk * NV + j) * 2) = p;
}

// ---------------------------------------------------------------------------
// Kernel 2: Jastrow quadratic form via f32 WMMA.
// One wave (32 lanes) per 16-batch tile. C = N_tile(16x128) * W(128x128),
// then y_b = sum_j C[b][j] * n[b][j].
//
// v_wmma_f32_16x16x4_f32 fragment layouts (ISA 7.12.2, wave32):
//   A (16x4, MxK):  lane L: M = L&15 ; VGPR0 = K = 2*(L>>4), VGPR1 = K+1
//   B (4x16, KxN):  lane L: N = L&15 ; VGPR0 = K = 2*(L>>4), VGPR1 = K+1
//   C (16x16):      lane L: N = L&15 ; VGPR v: M = v + 8*(L>>4)
// ---------------------------------------------------------------------------
__global__ __launch_bounds__(32) void jastrow_kernel(const float* __restrict__ n,
                                                     const float* __restrict__ Wp,
                                                     float* __restrict__ y) {
    const int b0   = blockIdx.x * 16;
    const int lane = threadIdx.x;      // 0..31
    const int m16  = lane & 15;
    const int hi   = lane >> 4;

    v8f acc[8] = {};                   // 8 column tiles of C (j0 = 16*jt)

    #pragma unroll 4
    for (int k0 = 0; k0 < NV; k0 += 4) {
        // A fragment: n[b0+m16][k0 + 2*hi + {0,1}], one b64 load
        v2f a = *(const v2f*)(n + (size_t)(b0 + m16) * NV + k0 + 2 * hi);
        // Packed-W row-pair index for this lane: kk = k0/2 + hi
        const float* wbase = Wp + (size_t)((k0 >> 1) + hi) * (NV * 2);
        #pragma unroll
        for (int jt = 0; jt < 8; ++jt) {
            // B fragment: {W[k][j], W[k+1][j]} in one b64 load
            v2f b = *(const v2f*)(wbase + (size_t)(jt * 16 + m16) * 2);
            acc[jt] = __builtin_amdgcn_wmma_f32_16x16x4_f32(
                false, a, false, b, (short)0, acc[jt], false, false);
        }
    }

    // y_b = sum_j C[b][j] * n[b][j]; lane holds rows v + 8*hi, col jt*16+m16
    float ypart[8];
    #pragma unroll
    for (int v = 0; v < 8; ++v) ypart[v] = 0.0f;

    #pragma unroll
    for (int jt = 0; jt < 8; ++jt) {
        #pragma unroll
        for (int v = 0; v < 8; ++v) {
            const int row = v + 8 * hi;
            float nv = n[(size_t)(b0 + row) * NV + jt * 16 + m16];
            ypart[v] += acc[jt][v] * nv;
        }
    }
    // reduce across the 16 lanes of the m16 dimension (stays within half-wave)
    #pragma unroll
    for (int off = 1; off < 16; off <<= 1) {
        #pragma unroll
        for (int v = 0; v < 8; ++v)
            ypart[v] += __shfl_xor(ypart[v], off, 32);
    }
    if (m16 == 0) {
        #pragma unroll
        for (int v = 0; v < 8; ++v)
            y[b0 + v + 8 * hi] = ypart[v];
    }
}

// ---------------------------------------------------------------------------
// Kernel 3: per-(batch,spin) 32x32 slogdet via partial-pivot LU in LDS.
// blockDim = 64: wave 0 -> spin 0, wave 1 -> spin 1 (wave32).
// Output: complex64 interleaved: out[2b] = sum(logabs) + y_b,
//                                out[2b+1] = pi * (#negative dets).
// ---------------------------------------------------------------------------
__global__ __launch_bounds__(64) void slogdet_kernel(const float* __restrict__ n,
                                                     const float* __restrict__ M,
                                                     const float* __restrict__ y,
                                                     float* __restrict__ out) {
    __shared__ float A[2][NF][NF + 1];   // stride 33: conflict-free on 64 banks
    __shared__ int   idxs[2][NF];
    __shared__ float r_log[2];
    __shared__ int   r_neg[2];

    const int b    = blockIdx.x;
    const int spin = threadIdx.x >> 5;
    const int lane = threadIdx.x & 31;

    // ---- rank-compaction of occupied orbitals (exactly NF per spin) ----
    const float* nb = n + (size_t)b * NV + spin * N_ORB;
    const float v0 = nb[lane];
    const float v1 = nb[lane + 32];
    const unsigned m0 = (unsigned)__ballot(v0 > 0.5f);
    const unsigned m1 = (unsigned)__ballot(v1 > 0.5f);
    const unsigned below = (lane == 0) ? 0u : (~0u >> (32 - lane));
    if (v0 > 0.5f) {
        int r = __popc(m0 & below);
        if (r < NF) idxs[spin][r] = lane;
    }
    if (v1 > 0.5f) {
        int r = __popc(m0) + __popc(m1 & below);
        if (r < NF) idxs[spin][r] = lane + 32;
    }
    __syncthreads();

    // ---- gather A = M[idx] (coalesced: lane = column, 128B per row) ----
    for (int r = 0; r < NF; ++r)
        A[spin][r][lane] = M[(size_t)idxs[spin][r] * NF + lane];
    __syncthreads();

    // ---- LU with partial pivoting; lane = row ----
    float logabs = 0.0f;
    int   neg    = 0;
    for (int k = 0; k < NF; ++k) {
        // argmax |A[r][k]| over rows r >= k (5-step shfl_xor reduction)
        float cv = (lane >= k) ? fabsf(A[spin][lane][k]) : -1.0f;
        int   p  = lane;
        #pragma unroll
        for (int off = 16; off > 0; off >>= 1) {
            float ov = __shfl_xor(cv, off, 32);
            int   op = __shfl_xor(p, off, 32);
            if (ov > cv || (ov == cv && op < p)) { cv = ov; p = op; }
        }
        if (p != k) {                     // row swap (lane = column)
            float t1 = A[spin][k][lane];
            float t2 = A[spin][p][lane];
            A[spin][k][lane] = t2;
            A[spin][p][lane] = t1;
            neg ^= 1;
        }
        __syncthreads();
        const float piv = A[spin][k][k];
        logabs += logf(fabsf(piv));
        if (piv < 0.0f) neg ^= 1;
        if (lane > k) {                   // eliminate; lane = row, serial cols
            float f = A[spin][lane][k] / piv;
            for (int j = k + 1; j < NF; ++j)
                A[spin][lane][j] -= f * A[spin][k][j];
        }
        __syncthreads();
    }

    if (lane == 0) { r_log[spin] = logabs; r_neg[spin] = neg; }
    __syncthreads();

    if (threadIdx.x == 0) {
        out[2 * b]     = r_log[0] + r_log[1] + y[b];
        out[2 * b + 1] = PI_F * (float)(r_neg[0] + r_neg[1]);
    }
}

// ---------------------------------------------------------------------------
extern "C" void kernel_launch(void* const* d_in, const int* in_sizes, int n_in,
                              void* d_out, int out_size, void* d_ws, size_t ws_size,
                              hipStream_t stream) {
    const float* n  = (const float*)d_in[0];   // (B, 128)
    const float* M  = (const float*)d_in[1];   // (64, 32)
    const float* jk = (const float*)d_in[2];   // (8128,)
    float* out = (float*)d_out;                // B complex64 -> 2B floats
    float* Wp  = (float*)d_ws;                 // packed W: 64*128 float2 = 16384 floats
    float* y   = Wp + NV * NV;                 // B floats

    build_w_kernel<<<NV / 2, NV, 0, stream>>>(jk, Wp);
    jastrow_kernel<<<BATCH / 16, 32, 0, stream>>>(n, Wp, y);
    slogdet_kernel<<<BATCH, 64, 0, stream>>>(n, M, y, out);
}